// SimpleProcessor_41437844471979
// MI455X (gfx1250) — compile-verified
//
#include <hip/hip_runtime.h>
#include <hip/hip_bf16.h>

#define N_NODES 100000
#define N_EDGES 1600000
#define FEAT    128
#define BATCH   2

typedef __attribute__((ext_vector_type(2))) float v2f;
typedef __attribute__((ext_vector_type(8))) float v8f;

// ---------------- degree / normalization ----------------
__global__ void k_deg_init(int* __restrict__ deg) {
    int n = blockIdx.x * blockDim.x + threadIdx.x;
    if (n < N_NODES) deg[n] = 1;                       // self-loop
}

__global__ void k_deg_count(const long long* __restrict__ ei, int* __restrict__ deg) {
    int e = blockIdx.x * blockDim.x + threadIdx.x;
    if (e < N_EDGES) {
        int d = (int)ei[2 * (long long)e + 1];
        atomicAdd(&deg[d], 1);
    }
}

__global__ void k_dinv(const int* __restrict__ deg, float* __restrict__ dinv) {
    int n = blockIdx.x * blockDim.x + threadIdx.x;
    if (n < N_NODES) dinv[n] = rsqrtf((float)deg[n]);
}

// ---------------- GEMM: hn = (x @ W) * dinv[row], out = (x @ W) * dinv[row]^2 ----
// Block: 256 threads = 8 waves. Block owns 16 rows; wave w owns cols [16w,16w+16).
__global__ __launch_bounds__(256) void k_gemm(const float* __restrict__ x,
                                              const float* __restrict__ W,
                                              const float* __restrict__ dinv,
                                              float* __restrict__ hn,
                                              float* __restrict__ out) {
    __shared__ float lds[16 * 132];                    // padded stride: bank-conflict free
    const int t    = threadIdx.x;
    const int row0 = blockIdx.x * 16;

    // Stage the 16x128 A tile: 512 float4, 2 per thread.
#pragma unroll
    for (int i = 0; i < 2; ++i) {
        int idx4 = t + i * 256;                        // 0..511
        int r    = idx4 >> 5;                          // 32 float4 per row
        int c4   = idx4 & 31;
        float4 v = *(const float4*)(x + (size_t)(row0 + r) * FEAT + c4 * 4);
        *(float4*)(&lds[r * 132 + c4 * 4]) = v;
    }
    __syncthreads();

    const int wave = t >> 5;
    const int lane = t & 31;
    const int m    = lane & 15;                        // A row / B-D column selector
    const int kh   = lane >> 4;                        // half-wave K shift
    const int col0 = wave * 16;
    const int n    = col0 + m;                         // output column owned by lane

    v8f acc = {};
#pragma unroll
    for (int kb = 0; kb < FEAT; kb += 4) {
        // A 16x4 (f32): lanes 0-15 K={kb,kb+1}, lanes 16-31 K={kb+2,kb+3}
        v2f a = *(const v2f*)(&lds[m * 132 + kb + 2 * kh]);
        // B 4x16 (f32): symmetric half-wave layout
        v2f b;
        b.x = W[(size_t)(kb + 2 * kh) * FEAT + n];
        b.y = W[(size_t)(kb + 2 * kh + 1) * FEAT + n];
        acc = __builtin_amdgcn_wmma_f32_16x16x4_f32(false, a, false, b,
                                                    (short)0, acc, false, false);
    }

    // Epilogue: D layout — VGPR v holds row (v + 8*kh), column n.
#pragma unroll
    for (int v = 0; v < 8; ++v) {
        int   row  = row0 + v + 8 * kh;
        int   node = row % N_NODES;                    // N_NODES % 16 == 0: no batch straddle
        float di   = dinv[node];
        float hval = acc[v] * di;                      // fold dinv[src] into hn
        size_t o   = (size_t)row * FEAT + n;
        hn[o]  = hval;
        out[o] = hval * di;                            // self-loop contribution
    }
}

// ---------------- edge scatter: one wave per (edge, batch) ----------------
__global__ __launch_bounds__(256) void k_scatter(const float* __restrict__ hn,
                                                 const float* __restrict__ dinv,
                                                 const long long* __restrict__ ei,
                                                 float* __restrict__ out) {
    int gid  = blockIdx.x * 256 + threadIdx.x;
    int wid  = gid >> 5;
    int lane = gid & 31;
    if (wid >= N_EDGES * BATCH) return;
    int b = wid & 1;
    int e = wid >> 1;

    int s = (int)ei[2 * (long long)e + 0];
    int d = (int)ei[2 * (long long)e + 1];
    float wgt = dinv[d];                               // dinv[src] already in hn

    const float4 v = *(const float4*)(hn + ((size_t)b * N_NODES + s) * FEAT + lane * 4);
    float* op = out + ((size_t)b * N_NODES + d) * FEAT + lane * 4;
    atomicAdd(op + 0, v.x * wgt);
    atomicAdd(op + 1, v.y * wgt);
    atomicAdd(op + 2, v.z * wgt);
    atomicAdd(op + 3, v.w * wgt);
}

// ---------------- bias + ReLU ----------------
__global__ void k_bias_relu(float* __restrict__ out, const float* __restrict__ bias) {
    int i = blockIdx.x * blockDim.x + threadIdx.x;     // float4 index
    const int total = BATCH * N_NODES * (FEAT / 4);
    if (i >= total) return;
    int g4 = i & (FEAT / 4 - 1);
    float4 bb = *(const float4*)(bias + g4 * 4);
    float4 v  = ((float4*)out)[i];
    v.x = fmaxf(v.x + bb.x, 0.0f);
    v.y = fmaxf(v.y + bb.y, 0.0f);
    v.z = fmaxf(v.z + bb.z, 0.0f);
    v.w = fmaxf(v.w + bb.w, 0.0f);
    ((float4*)out)[i] = v;
}

// ---------------- driver ----------------
static void run_layer(const float* x, const float* W, const float* bias,
                      const float* dinv, const long long* ei,
                      float* hn, float* out, hipStream_t stream) {
    k_gemm<<<(BATCH * N_NODES) / 16, 256, 0, stream>>>(x, W, dinv, hn, out);
    k_scatter<<<(N_EDGES * BATCH * 32) / 256, 256, 0, stream>>>(hn, dinv, ei, out);
    k_bias_relu<<<(BATCH * N_NODES * (FEAT / 4) + 255) / 256, 256, 0, stream>>>(out, bias);
}

extern "C" void kernel_launch(void* const* d_in, const int* in_sizes, int n_in,
                              void* d_out, int out_size, void* d_ws, size_t ws_size,
                              hipStream_t stream) {
    const float*     x0 = (const float*)d_in[0];
    const long long* ei = (const long long*)d_in[1];
    const float*     W1 = (const float*)d_in[2];
    const float*     b1 = (const float*)d_in[3];
    const float*     W2 = (const float*)d_in[4];
    const float*     b2 = (const float*)d_in[5];
    const float*     W3 = (const float*)d_in[6];
    const float*     b3 = (const float*)d_in[7];

    char* ws = (char*)d_ws;
    const size_t SZ = (size_t)BATCH * N_NODES * FEAT * sizeof(float);   // 102.4 MB
    float* hn   = (float*)(ws);
    float* x1   = (float*)(ws + SZ);
    int*   deg  = (int*)(ws + 2 * SZ);
    float* dinv = (float*)(ws + 2 * SZ + ((N_NODES * sizeof(int) + 255) & ~(size_t)255));
    float* outf = (float*)d_out;

    // symmetric GCN normalization (recomputed every call: deterministic)
    k_deg_init<<<(N_NODES + 255) / 256, 256, 0, stream>>>(deg);
    k_deg_count<<<(N_EDGES + 255) / 256, 256, 0, stream>>>(ei, deg);
    k_dinv<<<(N_NODES + 255) / 256, 256, 0, stream>>>(deg, dinv);

    run_layer(x0,   W1, b1, dinv, ei, hn, outf, stream);   // L1: in   -> d_out
    run_layer(outf, W2, b2, dinv, ei, hn, x1,   stream);   // L2: d_out-> ws.x1
    run_layer(x1,   W3, b3, dinv, ei, hn, outf, stream);   // L3: ws.x1-> d_out
}